// ReconGNN_10385230922556
// MI455X (gfx1250) — compile-verified
//
#include <hip/hip_runtime.h>
#include <hip/hip_bf16.h>

typedef __attribute__((ext_vector_type(2))) float v2f;
typedef __attribute__((ext_vector_type(8))) float v8f;

// ---------------------------------------------------------------------------
// Elementwise helpers
// ---------------------------------------------------------------------------
__global__ void zero_kernel(float* __restrict__ p, long long n) {
    long long i = (long long)blockIdx.x * blockDim.x + threadIdx.x;
    if (i < n) p[i] = 0.0f;
}

__global__ void deg_kernel(const int* __restrict__ col, float* __restrict__ deg, int E) {
    int e = blockIdx.x * blockDim.x + threadIdx.x;
    if (e < E) atomicAdd(&deg[col[e]], 1.0f);
}

__global__ void dinv_kernel(float* __restrict__ deg, int N) {
    int n = blockIdx.x * blockDim.x + threadIdx.x;
    if (n < N) {
        float d = deg[n];
        deg[n] = (d > 0.0f) ? rsqrtf(fmaxf(d, 1.0f)) : 0.0f;
    }
}

__global__ void edgew_kernel(const int* __restrict__ row, const int* __restrict__ col,
                             const float* __restrict__ dinv, float* __restrict__ w, int E) {
    int e = blockIdx.x * blockDim.x + threadIdx.x;
    if (e < E) w[e] = dinv[row[e]] * dinv[col[e]];
}

__global__ void relu_bias64_kernel(float* __restrict__ a, const float* __restrict__ b, long long n) {
    long long i = (long long)blockIdx.x * blockDim.x + threadIdx.x;
    if (i < n) a[i] = fmaxf(a[i] + b[(int)(i & 63)], 0.0f);
}

__global__ void bias40_kernel(float* __restrict__ a, const float* __restrict__ b, long long n) {
    long long i = (long long)blockIdx.x * blockDim.x + threadIdx.x;
    if (i < n) a[i] = a[i] + b[(int)(i % 40)];
}

// ---------------------------------------------------------------------------
// Dense GEMM via fp32 WMMA (V_WMMA_F32_16X16X4_F32), wave32.
//   C[N,Ncols] = X[N,K] * W^T, W is [Ncols,K] row-major.
// One wave computes one 16x16 C tile; K consumed in steps of 4.
// A lane layout (16x4 fp32 A): half=lane>>4, idx=lane&15
//   A: v2f {X[rb+idx][k0+2*half], X[rb+idx][k0+2*half+1]}
//   B: v2f {W[cb+idx][k0+2*half], W[cb+idx][k0+2*half+1]}
// C/D: acc[v] -> row rb + v + 8*half, col cb + idx.
// ---------------------------------------------------------------------------

// GEMM1: K=128, 64 output cols, 4 waves/block (one per col tile), 128 thr.
__global__ __launch_bounds__(128) void gemm1_wmma(const float* __restrict__ X,
                                                  const float* __restrict__ W1,
                                                  float* __restrict__ H1, int N) {
    const int wave = threadIdx.x >> 5;   // 0..3 -> column tile
    const int lane = threadIdx.x & 31;
    const int half = lane >> 4;
    const int idx  = lane & 15;
    const int rb   = blockIdx.x * 16;
    const int cb   = wave * 16;

    int arow = rb + idx;
    if (arow >= N) arow = N - 1;                 // clamp (row m of D only uses row m of A)
    const float* __restrict__ xrow = X  + (size_t)arow * 128;
    const float* __restrict__ wrow = W1 + (size_t)(cb + idx) * 128;

    v8f c = {};
#pragma unroll
    for (int k0 = 0; k0 < 128; k0 += 4) {
        const int kb = k0 + 2 * half;
        v2f a = *(const v2f*)(xrow + kb);
        v2f b = *(const v2f*)(wrow + kb);
        c = __builtin_amdgcn_wmma_f32_16x16x4_f32(false, a, false, b, (short)0, c, false, false);
    }

#pragma unroll
    for (int v = 0; v < 8; ++v) {
        int r = rb + v + 8 * half;
        if (r < N) H1[(size_t)r * 64 + (cb + idx)] = c[v];
    }
}

// GEMM2: K=64, 40 output cols -> 3 col tiles (last masked), 3 waves/block.
__global__ __launch_bounds__(96) void gemm2_wmma(const float* __restrict__ H2,
                                                 const float* __restrict__ W2,
                                                 float* __restrict__ H3, int N) {
    const int wave = threadIdx.x >> 5;   // 0..2 -> column tile
    const int lane = threadIdx.x & 31;
    const int half = lane >> 4;
    const int idx  = lane & 15;
    const int rb   = blockIdx.x * 16;
    const int cb   = wave * 16;

    int arow = rb + idx;
    if (arow >= N) arow = N - 1;
    const bool cvalid = (cb + idx) < 40;
    const float* __restrict__ xrow = H2 + (size_t)arow * 64;
    const float* __restrict__ wrow = W2 + (size_t)(cvalid ? (cb + idx) : 0) * 64;

    v8f c = {};
#pragma unroll
    for (int k0 = 0; k0 < 64; k0 += 4) {
        const int kb = k0 + 2 * half;
        v2f a = *(const v2f*)(xrow + kb);
        v2f b = *(const v2f*)(wrow + kb);
        if (!cvalid) { b.x = 0.0f; b.y = 0.0f; }   // mask padded B columns
        c = __builtin_amdgcn_wmma_f32_16x16x4_f32(false, a, false, b, (short)0, c, false, false);
    }

    if (cvalid) {
#pragma unroll
        for (int v = 0; v < 8; ++v) {
            int r = rb + v + 8 * half;
            if (r < N) H3[(size_t)r * 40 + (cb + idx)] = c[v];
        }
    }
}

// ---------------------------------------------------------------------------
// Edge scatter-add (gcn_conv aggregation). float4-vectorized; atomics resolve
// in L2 (entire destination fits in the 192MB L2).
// ---------------------------------------------------------------------------
__global__ void scatter64_kernel(const float* __restrict__ h,
                                 const int* __restrict__ row, const int* __restrict__ col,
                                 const float* __restrict__ w,
                                 float* __restrict__ out, int E) {
    long long t = (long long)blockIdx.x * blockDim.x + threadIdx.x;
    if (t >= (long long)E * 16) return;
    int e  = (int)(t >> 4);
    int dq = (int)(t & 15);
    int r  = row[e], c = col[e];
    float wt = w[e];
    const float4 v = *(const float4*)(h + (size_t)r * 64 + dq * 4);
    float* o = out + (size_t)c * 64 + dq * 4;
    atomicAdd(o + 0, v.x * wt);
    atomicAdd(o + 1, v.y * wt);
    atomicAdd(o + 2, v.z * wt);
    atomicAdd(o + 3, v.w * wt);
}

__global__ void scatter40_kernel(const float* __restrict__ h,
                                 const int* __restrict__ row, const int* __restrict__ col,
                                 const float* __restrict__ w,
                                 float* __restrict__ out, int E) {
    long long t = (long long)blockIdx.x * blockDim.x + threadIdx.x;
    if (t >= (long long)E * 10) return;
    int e  = (int)(t / 10);
    int dq = (int)(t % 10);
    int r  = row[e], c = col[e];
    float wt = w[e];
    const float4 v = *(const float4*)(h + (size_t)r * 40 + dq * 4);   // 40*4B=160B -> 16B aligned
    float* o = out + (size_t)c * 40 + dq * 4;
    atomicAdd(o + 0, v.x * wt);
    atomicAdd(o + 1, v.y * wt);
    atomicAdd(o + 2, v.z * wt);
    atomicAdd(o + 3, v.w * wt);
}

// ---------------------------------------------------------------------------
// Host launcher
// ---------------------------------------------------------------------------
static inline long long align_up(long long x, long long a) { return (x + a - 1) / a * a; }

extern "C" void kernel_launch(void* const* d_in, const int* in_sizes, int n_in,
                              void* d_out, int out_size, void* d_ws, size_t ws_size,
                              hipStream_t stream) {
    const float* x   = (const float*)d_in[0];
    const int*   ei  = (const int*)d_in[1];
    const float* W1  = (const float*)d_in[2];
    const float* b1  = (const float*)d_in[3];
    const float* W2  = (const float*)d_in[4];
    const float* b2  = (const float*)d_in[5];
    float* out = (float*)d_out;

    const int N = in_sizes[0] / 128;
    const int E = in_sizes[1] / 2;
    const int* row = ei;        // edge_index[0]
    const int* col = ei + E;    // edge_index[1]

    // Workspace layout (floats), 256B-aligned segments
    float* ws = (float*)d_ws;
    long long off = 0;
    float* dinv = ws + off; off = align_up(off + N, 64);
    float* ew   = ws + off; off = align_up(off + E, 64);
    float* h1   = ws + off; off = align_up(off + (long long)N * 64, 64);
    float* agg  = ws + off; off = align_up(off + (long long)N * 64, 64);
    float* h3   = ws + off; off = align_up(off + (long long)N * 40, 64);
    (void)ws_size; (void)n_in; (void)out_size;

    const int T = 256;
    const int rowTiles = (N + 15) / 16;

    // degrees -> d^-1/2 -> edge weights (shared by both conv layers)
    zero_kernel<<<(N + T - 1) / T, T, 0, stream>>>(dinv, N);
    deg_kernel<<<(E + T - 1) / T, T, 0, stream>>>(col, dinv, E);
    dinv_kernel<<<(N + T - 1) / T, T, 0, stream>>>(dinv, N);
    edgew_kernel<<<(E + T - 1) / T, T, 0, stream>>>(row, col, dinv, ew, E);

    // layer 1: h1 = x @ W1^T  (fp32 WMMA)
    gemm1_wmma<<<rowTiles, 128, 0, stream>>>(x, W1, h1, N);

    // conv1: agg = scatter_add(h1[row]*w -> col)
    long long n64 = (long long)N * 64;
    zero_kernel<<<(int)((n64 + T - 1) / T), T, 0, stream>>>(agg, n64);
    long long w1n = (long long)E * 16;
    scatter64_kernel<<<(int)((w1n + T - 1) / T), T, 0, stream>>>(h1, row, col, ew, agg, E);

    // relu(agg + b1) in place
    relu_bias64_kernel<<<(int)((n64 + T - 1) / T), T, 0, stream>>>(agg, b1, n64);

    // layer 2: h3 = agg @ W2^T  (fp32 WMMA)
    gemm2_wmma<<<rowTiles, 96, 0, stream>>>(agg, W2, h3, N);

    // conv2: out = scatter_add(h3[row]*w -> col) + b2
    long long n40 = (long long)N * 40;
    zero_kernel<<<(int)((n40 + T - 1) / T), T, 0, stream>>>(out, n40);
    long long w2n = (long long)E * 10;
    scatter40_kernel<<<(int)((w2n + T - 1) / T), T, 0, stream>>>(h3, row, col, ew, out, E);
    bias40_kernel<<<(int)((n40 + T - 1) / T), T, 0, stream>>>(out, b2, n40);
}